// GMSDR_55989193670747
// MI455X (gfx1250) — compile-verified
//
#include <hip/hip_runtime.h>

// ---------------- problem constants ----------------
#define NN 883          // nodes
#define NP 896          // nodes padded to multiple of 16 (7*128)
#define DD 64           // rnn units
#define BB 32           // batch
#define LL 12           // encoder steps
#define HH 12           // decoder steps
#define PREK 6          // pre_k
#define FF 128          // gconv input features = D + PRE_V*D
#define XC (BB*FF)      // 4096 cols of x buffers
#define HC (BB*DD)      // 2048 cols of hidden buffers
#define LRELU 0.01f     // jax.nn.leaky_relu default slope

#define LDSPITCH 20     // LDS row pitch (dwords): 20*m mod 64 -> 16 distinct banks

typedef __attribute__((ext_vector_type(2))) float v2f;
typedef __attribute__((ext_vector_type(8))) float v8f;

// =====================================================================
// Unified f32 WMMA GEMM:  C = alpha * (A @ B) + beta * E   (per batch z)
//   A: (M x K) lda, batch stride aB     (M = gridDim.x*128, padded)
//   B: (K x N) ldb, batch stride bB     (N = gridDim.y*64)
//   C: ldc / cB,  E: lde / eB (may alias C for accumulation)
// Block: 256 threads = 8 waves; tile 128(M) x 64(N); K staged by 16 in LDS.
// Wave w = (wm = w&3, wn = w>>2) computes a 32x32 region: 2x2 16x16 tiles,
// i.e. 4 WMMAs per fragment set, 16 WMMAs per K-stage.
// LDS layout [row][k] pitch 20 -> fragments are single aligned v2f loads.
// =====================================================================
__launch_bounds__(256)
__global__ void gemm_wmma(const float* __restrict__ A, int lda, long aB,
                          const float* __restrict__ B, int ldb, long bB,
                          float* __restrict__ C, int ldc, long cB,
                          const float* __restrict__ E, int lde, long eB,
                          float alpha, float beta, int K) {
  __shared__ float As[128 * LDSPITCH];   // [m][k], 10 KB
  __shared__ float Bs[64 * LDSPITCH];    // [n][k], 5 KB

  const int tid  = threadIdx.x;
  const int lane = tid & 31;     // wave32
  const int wid  = tid >> 5;
  const int wm   = wid & 3;      // M group (32 rows)
  const int wn   = wid >> 2;     // N group (32 cols)

  A += (long)blockIdx.z * aB + (long)blockIdx.x * 128 * lda;
  B += (long)blockIdx.z * bB + (long)blockIdx.y * 64;
  C += (long)blockIdx.z * cB + (long)blockIdx.x * 128 * ldc + (long)blockIdx.y * 64;
  if (E) E += (long)blockIdx.z * eB + (long)blockIdx.x * 128 * lde + (long)blockIdx.y * 64;

  v8f acc00 = {0.f,0.f,0.f,0.f,0.f,0.f,0.f,0.f};
  v8f acc01 = acc00, acc10 = acc00, acc11 = acc00;

  // A staging: 128 rows x 16 k = 2048 floats; thread -> row t>>1, 8 floats
  const int ar = tid >> 1;
  const int ac = (tid & 1) * 8;
  // B staging: 16 k x 64 n = 1024 floats; thread -> k row t>>4, 4 cols
  const int br = tid >> 4;
  const int bc = (tid & 15) * 4;

  const int ml   = lane & 15;
  const int koff = (lane < 16) ? 0 : 2;   // lanes 16-31 hold K+2,K+3

  for (int k0 = 0; k0 < K; k0 += 16) {
    float4 av0 = *(const float4*)(A + (long)ar * lda + k0 + ac);
    float4 av1 = *(const float4*)(A + (long)ar * lda + k0 + ac + 4);
    float4 bv  = *(const float4*)(B + (long)(k0 + br) * ldb + bc);
    if (k0 + 16 < K) {   // prefetch next stage into near caches
      __builtin_prefetch(A + (long)ar * lda + k0 + 16 + ac, 0, 3);
      __builtin_prefetch(B + (long)(k0 + 16 + br) * ldb + bc, 0, 3);
    }
    __syncthreads();     // previous stage fully consumed
    *(float4*)&As[ar * LDSPITCH + ac]     = av0;
    *(float4*)&As[ar * LDSPITCH + ac + 4] = av1;
    Bs[(bc + 0) * LDSPITCH + br] = bv.x;   // transpose B: [n][k]
    Bs[(bc + 1) * LDSPITCH + br] = bv.y;
    Bs[(bc + 2) * LDSPITCH + br] = bv.z;
    Bs[(bc + 3) * LDSPITCH + br] = bv.w;
    __syncthreads();

#pragma unroll
    for (int kk = 0; kk < 4; ++kk) {
      const int kc = kk * 4 + koff;
      v2f a0 = *(const v2f*)&As[(wm * 32      + ml) * LDSPITCH + kc];
      v2f a1 = *(const v2f*)&As[(wm * 32 + 16 + ml) * LDSPITCH + kc];
      v2f b0 = *(const v2f*)&Bs[(wn * 32      + ml) * LDSPITCH + kc];
      v2f b1 = *(const v2f*)&Bs[(wn * 32 + 16 + ml) * LDSPITCH + kc];
      acc00 = __builtin_amdgcn_wmma_f32_16x16x4_f32(false, a0, false, b0, (short)0, acc00, false, false);
      acc01 = __builtin_amdgcn_wmma_f32_16x16x4_f32(false, a0, false, b1, (short)0, acc01, false, false);
      acc10 = __builtin_amdgcn_wmma_f32_16x16x4_f32(false, a1, false, b0, (short)0, acc10, false, false);
      acc11 = __builtin_amdgcn_wmma_f32_16x16x4_f32(false, a1, false, b1, (short)0, acc11, false, false);
    }
  }

  // C/D layout: VGPR r -> row r (lanes 0-15) / row 8+r (lanes 16-31), col = lane&15
  const int cn    = lane & 15;
  const int rbase = (lane < 16) ? 0 : 8;
#pragma unroll
  for (int r = 0; r < 8; ++r) {
    const int row0 = wm * 32 + rbase + r;
    const int row1 = row0 + 16;
    const int col0 = wn * 32 + cn;
    const int col1 = col0 + 16;
    float v00 = alpha * acc00[r], v01 = alpha * acc01[r];
    float v10 = alpha * acc10[r], v11 = alpha * acc11[r];
    if (beta != 0.f) {
      v00 += beta * E[(long)row0 * lde + col0];
      v01 += beta * E[(long)row0 * lde + col1];
      v10 += beta * E[(long)row1 * lde + col0];
      v11 += beta * E[(long)row1 * lde + col1];
    }
    C[(long)row0 * ldc + col0] = v00;
    C[(long)row0 * ldc + col1] = v01;
    C[(long)row1 * ldc + col0] = v10;
    C[(long)row1 * ldc + col1] = v11;
  }
}

// ---------------- support-matrix preparation ----------------
__global__ void k_rowcol(const float* __restrict__ A, float* rs, float* cs) {
  __shared__ float sr[256], sc[256];
  int j = blockIdx.x, t = threadIdx.x;
  float r = 0.f, c = 0.f;
  for (int m = t; m < NN; m += 256) { r += A[j*NN + m]; c += A[m*NN + j]; }
  sr[t] = r; sc[t] = c; __syncthreads();
  for (int o = 128; o > 0; o >>= 1) {
    if (t < o) { sr[t] += sr[t+o]; sc[t] += sc[t+o]; }
    __syncthreads();
  }
  if (t == 0) { rs[j] = sr[0]; cs[j] = sc[0]; }
}

// P1[m,n] = A[n,m]/rowsum(n) ; P2[m,n] = A[m,n]/colsum(n); zero in padding
__global__ void k_build_p(const float* __restrict__ A, const float* rs, const float* cs,
                          float* __restrict__ P1, float* __restrict__ P2) {
  int idx = blockIdx.x * 256 + threadIdx.x;
  if (idx >= NP * NP) return;
  int mm = idx / NP, n = idx % NP;
  float p1 = 0.f, p2 = 0.f;
  if (mm < NN && n < NN) { p1 = A[n*NN + mm] / rs[n]; p2 = A[mm*NN + n] / cs[n]; }
  P1[idx] = p1; P2[idx] = p2;
}

__global__ void k_zero(float* p, long n) {
  long i = (long)blockIdx.x * 256 + threadIdx.x;
  if (i < n) p[i] = 0.f;
}

__global__ void k_fillbias(float* __restrict__ conv, const float* __restrict__ bias) {
  int idx = blockIdx.x * 256 + threadIdx.x;      // NP*HC exact
  conv[idx] = bias[idx & (DD - 1)];
}

// x[n][b*128+f] : f<64 -> input features, f>=64 -> newest hidden state
// mode 0: src = history (B,L,N,1), project with mlp ; mode 1: src = enc_out (NP x HC)
__global__ void k_build_x(float* __restrict__ x, const float* __restrict__ hprev,
                          const float* __restrict__ src, const float* __restrict__ mlpw,
                          const float* __restrict__ mlpb, int mode, int l) {
  int idx = blockIdx.x * 256 + threadIdx.x;      // NP*XC exact
  int n = idx >> 12, bf = idx & 4095;
  int b = bf >> 7, f = bf & 127;
  float v;
  if (f < DD) {
    if (mode == 0) v = (n < NN) ? src[(b * LL + l) * NN + n] * mlpw[f] + mlpb[f] : 0.f;
    else           v = src[n * HC + b * DD + f];
  } else {
    v = hprev[n * HC + b * DD + (f - DD)];
  }
  x[idx] = v;
}

// scores[b,k] = sum_{n<NN,d} (hxk[k] + R[k]) * attW  + att_b
__global__ void k_scores(const float* h0, const float* h1, const float* h2,
                         const float* h3, const float* h4, const float* h5,
                         const float* __restrict__ R, const float* __restrict__ aw,
                         const float* __restrict__ ab, float* __restrict__ scores) {
  const float* hs[PREK] = {h0, h1, h2, h3, h4, h5};
  int b = blockIdx.x / PREK, k = blockIdx.x % PREK;
  const float* h  = hs[k];
  const float* Rk = R + (long)k * NN * DD;
  int t = threadIdx.x;
  float s = 0.f;
  for (int i = t; i < NN * DD; i += 256) {
    int n = i >> 6, d = i & 63;
    s += (h[n * HC + b * DD + d] + Rk[i]) * aw[i];
  }
  __shared__ float red[256];
  red[t] = s; __syncthreads();
  for (int o = 128; o > 0; o >>= 1) {
    if (t < o) red[t] += red[t + o];
    __syncthreads();
  }
  if (t == 0) scores[b * PREK + k] = red[0] + ab[0];
}

__global__ void k_softmax(const float* __restrict__ scores, float* __restrict__ w) {
  int b = threadIdx.x;
  if (b >= BB) return;
  float s[PREK], m = -1e30f;
  for (int k = 0; k < PREK; ++k) { s[k] = scores[b * PREK + k]; m = fmaxf(m, s[k]); }
  float sum = 0.f;
  for (int k = 0; k < PREK; ++k) { s[k] = __expf(s[k] - m); sum += s[k]; }
  for (int k = 0; k < PREK; ++k) w[b * PREK + k] = s[k] / sum;
}

// cell epilogue: leaky_relu(conv), output = (h5+R5)+conv, att = sum_k (hk+Rk)*w
__global__ void k_cellout(const float* h0, const float* h1, const float* h2,
                          const float* h3, const float* h4, const float* h5,
                          const float* __restrict__ conv, const float* __restrict__ R,
                          const float* __restrict__ w, float* __restrict__ newH,
                          float* __restrict__ outb) {
  int idx = blockIdx.x * 256 + threadIdx.x;      // NP*HC exact
  int n = idx >> 11, bd = idx & 2047;
  int b = bd >> 6, d = bd & 63;
  const float* hs[PREK] = {h0, h1, h2, h3, h4, h5};
  float c = conv[idx];
  c = (c > 0.f) ? c : LRELU * c;
  float att = 0.f, ns5 = 0.f;
#pragma unroll
  for (int k = 0; k < PREK; ++k) {
    float rv = (n < NN) ? R[((long)k * NN + n) * DD + d] : 0.f;
    float ns = hs[k][idx] + rv;
    att += ns * w[b * PREK + k];
    if (k == PREK - 1) ns5 = ns;
  }
  float output = ns5 + c;
  newH[idx] = output;           // becomes new hx_k[PREK-1]
  outb[idx] = output + att;     // cell output
}

__global__ void k_proj(const float* __restrict__ outb, const float* __restrict__ pw,
                       const float* __restrict__ pb, float* __restrict__ out, int t) {
  int i = blockIdx.x * 256 + threadIdx.x;
  if (i >= BB * NN) return;
  int b = i / NN, n = i % NN;
  const float* p = outb + (long)n * HC + b * DD;
  float s = 0.f;
  for (int d = 0; d < DD; ++d) s += p[d] * pw[d];
  out[((long)b * HH + t) * NN + n] = s + pb[0];
}

// =====================================================================
extern "C" void kernel_launch(void* const* d_in, const int* in_sizes, int n_in,
                              void* d_out, int out_size, void* d_ws, size_t ws_size,
                              hipStream_t stream) {
  const float* hist  = (const float*)d_in[0];
  const float* adj   = (const float*)d_in[1];
  const float* emlpw = (const float*)d_in[2];
  const float* emlpb = (const float*)d_in[3];
  const float* egw   = (const float*)d_in[4];
  const float* egb   = (const float*)d_in[5];
  const float* eR    = (const float*)d_in[6];
  const float* eaw   = (const float*)d_in[7];
  const float* eab   = (const float*)d_in[8];
  const float* dgw   = (const float*)d_in[9];
  const float* dgb   = (const float*)d_in[10];
  const float* dR    = (const float*)d_in[11];
  const float* daw   = (const float*)d_in[12];
  const float* dab   = (const float*)d_in[13];
  const float* pw    = (const float*)d_in[14];
  const float* pb    = (const float*)d_in[15];
  float* out = (float*)d_out;

  // ---- workspace layout (~196 MB, all f32) ----
  float* ws = (float*)d_ws;
  size_t off = 0;
  auto alloc = [&](size_t n) { float* p = ws + off; off += (n + 3) & ~(size_t)3; return p; };
  float* P1      = alloc((size_t)NP * NP);
  float* P2      = alloc((size_t)NP * NP);
  float* rs      = alloc(NP);
  float* cs      = alloc(NP);
  float* xbuf    = alloc((size_t)NP * XC);
  float* t1      = alloc((size_t)NP * XC);
  float* t2      = alloc((size_t)NP * XC);
  float* ring    = alloc((size_t)7 * NP * HC);   // hx_k ring (6 live + 1 incoming)
  float* conv    = alloc((size_t)NP * HC);
  float* encouts = alloc((size_t)LL * NP * HC);
  float* decoutb = alloc((size_t)NP * HC);
  float* scores  = alloc(256);
  float* wts     = alloc(256);
  (void)ws_size; (void)in_sizes; (void)n_in; (void)out_size;

  // ---- supports ----
  k_rowcol <<<NN, 256, 0, stream>>>(adj, rs, cs);
  k_build_p<<<(NP * NP + 255) / 256, 256, 0, stream>>>(adj, rs, cs, P1, P2);
  long ringN = (long)7 * NP * HC;
  k_zero<<<(unsigned)((ringN + 255) / 256), 256, 0, stream>>>(ring, ringN);

  int rb = 0;
  auto Hk = [&](int k) { return ring + (size_t)((rb + k) % 7) * NP * HC; };

  auto gemm = [&](const float* A, int lda, long aB, const float* B, int ldb, long bB,
                  float* C, int ldc, long cB, const float* E, int lde, long eB,
                  float alpha, float beta, int K, int mt, int nt, int bz) {
    gemm_wmma<<<dim3(mt, nt, bz), 256, 0, stream>>>(A, lda, aB, B, ldb, bB,
                                                    C, ldc, cB, E, lde, eB,
                                                    alpha, beta, K);
  };

  for (int step = 0; step < LL + HH; ++step) {
    bool enc = step < LL;
    const float* gw = enc ? egw : dgw;
    const float* gb = enc ? egb : dgb;
    const float* R  = enc ? eR  : dR;
    const float* aw = enc ? eaw : daw;
    const float* ab = enc ? eab : dab;

    // assemble x = [proj(input) | newest hidden]
    if (enc)
      k_build_x<<<NP * XC / 256, 256, 0, stream>>>(xbuf, Hk(PREK - 1), hist,
                                                   emlpw, emlpb, 0, step);
    else
      k_build_x<<<NP * XC / 256, 256, 0, stream>>>(xbuf, Hk(PREK - 1),
                                                   encouts + (size_t)(step - LL) * NP * HC,
                                                   emlpw, emlpb, 1, 0);

    // conv = bias, then accumulate 5 diffusion terms: conv += X_mat @ W_mat
    k_fillbias<<<NP * HC / 256, 256, 0, stream>>>(conv, gb);
    auto gconv = [&](const float* X, int mat) {
      gemm(X, XC, 128, gw + mat * DD, 5 * DD, 0,
           conv, HC, DD, conv, HC, DD,
           1.f, 1.f, FF, NP / 128, 1, BB);
    };

    gconv(xbuf, 0);
    // x1 = P1 @ x
    gemm(P1, NP, 0, xbuf, XC, 0, t1, XC, 0, nullptr, 0, 0, 1.f,  0.f, NP, NP/128, XC/64, 1);
    gconv(t1, 1);
    // x2 = 2*P1@x1 - x
    gemm(P1, NP, 0, t1,   XC, 0, t2, XC, 0, xbuf, XC, 0, 2.f, -1.f, NP, NP/128, XC/64, 1);
    gconv(t2, 2);
    // x3 = P2 @ x
    gemm(P2, NP, 0, xbuf, XC, 0, t1, XC, 0, nullptr, 0, 0, 1.f,  0.f, NP, NP/128, XC/64, 1);
    gconv(t1, 3);
    // x4 = 2*P2@x3 - x
    gemm(P2, NP, 0, t1,   XC, 0, t2, XC, 0, xbuf, XC, 0, 2.f, -1.f, NP, NP/128, XC/64, 1);
    gconv(t2, 4);

    // attention over pre_k states
    k_scores <<<BB * PREK, 256, 0, stream>>>(Hk(0), Hk(1), Hk(2), Hk(3), Hk(4), Hk(5),
                                             R, aw, ab, scores);
    k_softmax<<<1, 32, 0, stream>>>(scores, wts);

    float* outb = enc ? (encouts + (size_t)step * NP * HC) : decoutb;
    float* newH = ring + (size_t)((rb + 6) % 7) * NP * HC;
    k_cellout<<<NP * HC / 256, 256, 0, stream>>>(Hk(0), Hk(1), Hk(2), Hk(3), Hk(4), Hk(5),
                                                 conv, R, wts, newH, outb);
    if (!enc)
      k_proj<<<(BB * NN + 255) / 256, 256, 0, stream>>>(decoutb, pw, pb, out, step - LL);

    rb = (rb + 1) % 7;   // shift hx_k
  }
}